// Tokenizer_34668976013865
// MI455X (gfx1250) — compile-verified
//
#include <hip/hip_runtime.h>

#define N_NODES    10000
#define N_EDGES    640000
#define EMB        128
#define NUM_LAYERS 4
#define STATS_BLOCKS 50

typedef __attribute__((ext_vector_type(2))) float v2f;
typedef __attribute__((ext_vector_type(8))) float v8f;

// ---------------- CSR build (once per call; edges are layer-invariant) ------

__global__ void zero_i32(int* __restrict__ p, int n) {
  int i = blockIdx.x * blockDim.x + threadIdx.x;
  if (i < n) p[i] = 0;
}

__global__ void count_deg(const int* __restrict__ dst, int* __restrict__ deg, int nE) {
  int i = blockIdx.x * blockDim.x + threadIdx.x;
  if (i < nE) atomicAdd(&deg[dst[i]], 1);
}

// Single-block chunked Hillis-Steele exclusive scan over N_NODES degrees.
__global__ void scan_offsets(const int* __restrict__ deg, int* __restrict__ offsets,
                             int* __restrict__ cursor, int n) {
  __shared__ int lds[1024];
  __shared__ int carryS;
  int t = threadIdx.x;
  if (t == 0) carryS = 0;
  __syncthreads();
  for (int base = 0; base < n; base += 1024) {
    int i = base + t;
    int val = (i < n) ? deg[i] : 0;
    lds[t] = val;
    __syncthreads();
    for (int off = 1; off < 1024; off <<= 1) {
      int add = (t >= off) ? lds[t - off] : 0;
      __syncthreads();
      lds[t] += add;
      __syncthreads();
    }
    int carry = carryS;
    int excl  = carry + lds[t] - val;
    if (i < n) { offsets[i] = excl; cursor[i] = excl; }
    __syncthreads();
    if (t == 1023) carryS = carry + lds[1023];
    __syncthreads();
  }
  if (t == 0) offsets[n] = carryS;
}

__global__ void fill_edges(const int* __restrict__ src, const int* __restrict__ dst,
                           int* __restrict__ cursor, int* __restrict__ eidx, int nE) {
  int i = blockIdx.x * blockDim.x + threadIdx.x;
  if (i < nE) {
    int p = atomicAdd(&cursor[dst[i]], 1);
    eidx[p] = src[i];
  }
}

// ---------------- GIN aggregation: one wave32 per node, float4 per lane -----

__global__ void gin_gather(const float* __restrict__ hin, const int* __restrict__ offsets,
                           const int* __restrict__ eidx, float* __restrict__ acc, int nNodes) {
  int wave = (int)((blockIdx.x * blockDim.x + threadIdx.x) >> 5);
  int lane = threadIdx.x & 31;
  if (wave >= nNodes) return;
  const float4* h4 = (const float4*)hin;
  float4 a = h4[wave * 32 + lane];          // (1+eps)*h self term, eps=0
  int e0 = offsets[wave];
  int e1 = offsets[wave + 1];
  for (int e = e0; e < e1; e += 32) {
    int me  = e + lane;
    int sid = (me < e1) ? eidx[me] : 0;     // cooperative edge-id fetch
    int cnt = e1 - e; if (cnt > 32) cnt = 32;
    #pragma unroll 4
    for (int j = 0; j < cnt; ++j) {
      int s = __shfl(sid, j, 32);
      float4 b = h4[s * 32 + lane];
      a.x += b.x; a.y += b.y; a.z += b.z; a.w += b.w;
    }
  }
  ((float4*)acc)[wave * 32 + lane] = a;
}

// ---------------- BN column stats via V_WMMA_F32_16X16X4_F32 ----------------
// A = ones(16x4)  =>  C[i][j] += sum_k B[k][j]: every C row carries the column
// sums. Wave w of the block owns channel tile [16w, 16w+16). Per 4-row step,
// lanes 0-15 supply rows r,r+1 and lanes 16-31 rows r+2,r+3 for column lane%16.
__global__ void bn_stats_wmma(const float* __restrict__ h, float* __restrict__ partials,
                              int nRows, int rowsPerBlk) {
  int waveId  = threadIdx.x >> 5;       // 0..7 -> channel tile
  int lane    = threadIdx.x & 31;
  int colBase = waveId * 16;
  int n       = colBase + (lane & 15);
  int half    = lane >> 4;
  int rStart  = blockIdx.x * rowsPerBlk;
  int rEnd    = rStart + rowsPerBlk; if (rEnd > nRows) rEnd = nRows;

  v2f ones; ones.x = 1.0f; ones.y = 1.0f;
  v8f cs = {};   // column sums
  v8f cq = {};   // column sums of squares
  for (int r = rStart; r < rEnd; r += 4) {
    int r0 = r + half * 2;
    int r1 = r0 + 1;
    float b0 = (r0 < nRows) ? h[r0 * EMB + n] : 0.0f;
    float b1 = (r1 < nRows) ? h[r1 * EMB + n] : 0.0f;
    v2f b;  b.x  = b0;       b.y  = b1;
    v2f b2; b2.x = b0 * b0;  b2.y = b1 * b1;
    cs = __builtin_amdgcn_wmma_f32_16x16x4_f32(false, ones, false, b,  (short)0, cs, false, false);
    cq = __builtin_amdgcn_wmma_f32_16x16x4_f32(false, ones, false, b2, (short)0, cq, false, false);
  }
  // C VGPR0: lane l holds (M = 8*(l>>4), N = l&15); all rows equal the col sum.
  float* pb = partials + blockIdx.x * 256;
  if (lane < 16) pb[colBase + lane]              = cs[0];
  else           pb[128 + colBase + (lane - 16)] = cq[0];
}

// Fixed-order partial reduction + scale/shift precompute (deterministic).
__global__ void bn_finalize(const float* __restrict__ partials, int nPart,
                            const float* __restrict__ gamma, const float* __restrict__ beta,
                            float* __restrict__ scsh, int layer, float invN) {
  int c = threadIdx.x;  // 0..127
  float s = 0.0f, sq = 0.0f;
  for (int p = 0; p < nPart; ++p) {
    s  += partials[p * 256 + c];
    sq += partials[p * 256 + 128 + c];
  }
  float mu   = s * invN;
  float var  = sq * invN - mu * mu;
  if (var < 0.0f) var = 0.0f;
  float rstd = rsqrtf(var + 1e-5f);
  float sc   = rstd * gamma[layer * EMB + c];
  scsh[c]       = sc;
  scsh[128 + c] = beta[layer * EMB + c] - mu * sc;
}

// In-place normalize: h = h*scale[c] + shift[c], float4-vectorized.
__global__ void bn_apply(float4* __restrict__ h, const float4* __restrict__ sc4,
                         const float4* __restrict__ sh4, int total4) {
  int i = blockIdx.x * blockDim.x + threadIdx.x;
  if (i >= total4) return;
  float4 v = h[i];
  float4 s = sc4[i & 31];
  float4 t = sh4[i & 31];
  v.x = fmaf(v.x, s.x, t.x);
  v.y = fmaf(v.y, s.y, t.y);
  v.z = fmaf(v.z, s.z, t.z);
  v.w = fmaf(v.w, s.w, t.w);
  h[i] = v;
}

// ---------------------------------------------------------------------------

extern "C" void kernel_launch(void* const* d_in, const int* in_sizes, int n_in,
                              void* d_out, int out_size, void* d_ws, size_t ws_size,
                              hipStream_t stream) {
  const float* x     = (const float*)d_in[0];
  const float* gamma = (const float*)d_in[1];
  const float* beta  = (const float*)d_in[2];
  const int*   src   = (const int*)d_in[3];
  const int*   dst   = (const int*)d_in[4];
  float* out = (float*)d_out;

  char* ws = (char*)d_ws;
  float* hbuf    = (float*)(ws);                 // 10000*128 f32 = 5,120,000 B
  int*   eidx    = (int*)  (ws + 5120000);       // 640000 i32   = 2,560,000 B
  int*   deg     = (int*)  (ws + 7680000);       // 10000 i32
  int*   offsets = (int*)  (ws + 7720064);       // 10001 i32
  int*   cursor  = (int*)  (ws + 7760128);       // 10000 i32
  float* parts   = (float*)(ws + 7800192);       // STATS_BLOCKS*256 f32
  float* scsh    = (float*)(ws + 7851392);       // 256 f32

  // CSR build (edges constant across layers)
  zero_i32   <<<(N_NODES + 255) / 256, 256, 0, stream>>>(deg, N_NODES);
  count_deg  <<<(N_EDGES + 255) / 256, 256, 0, stream>>>(dst, deg, N_EDGES);
  scan_offsets<<<1, 1024, 0, stream>>>(deg, offsets, cursor, N_NODES);
  fill_edges <<<(N_EDGES + 255) / 256, 256, 0, stream>>>(src, dst, cursor, eidx, N_EDGES);

  int rowsPerBlk = ((N_NODES + STATS_BLOCKS - 1) / STATS_BLOCKS + 3) & ~3;

  for (int l = 0; l < NUM_LAYERS; ++l) {
    const float* hin = (l == 0) ? x : ((l & 1) ? hbuf : out);
    float*       acc = (l & 1) ? out : hbuf;     // layer 3 -> d_out

    gin_gather   <<<(N_NODES * 32 + 255) / 256, 256, 0, stream>>>(hin, offsets, eidx, acc, N_NODES);
    bn_stats_wmma<<<STATS_BLOCKS, 256, 0, stream>>>(acc, parts, N_NODES, rowsPerBlk);
    bn_finalize  <<<1, 128, 0, stream>>>(parts, STATS_BLOCKS, gamma, beta, scsh, l, 1.0f / N_NODES);
    bn_apply     <<<(N_NODES * 32 + 255) / 256, 256, 0, stream>>>((float4*)acc,
                     (const float4*)scsh, (const float4*)(scsh + 128), N_NODES * 32);
  }
}